// STA_73650099191827
// MI455X (gfx1250) — compile-verified
//
#include <hip/hip_runtime.h>

// Problem constants (fixed by the reference)
#define LL 16
#define NBATCH 4
#define NN 4096
#define KDIM 2048

typedef float v2f __attribute__((ext_vector_type(2)));
typedef float v8f __attribute__((ext_vector_type(8)));

// CDNA5 native fp32 WMMA: D(16x16) = A(16x4) * B(4x16) + C
static __device__ __forceinline__ v8f wmma4(v2f a, v2f b, v8f c) {
    return __builtin_amdgcn_wmma_f32_16x16x4_f32(
        false, a, false, b, (short)0, c, false, false);
}

// ---------------------------------------------------------------------------
// Kernel 1: Xe[b,m,n] = sum_l S[b,l,n]*W1x[l,m];  Se[b,m,n] = sum_l X[b,l,n]*W1s[l,m]
// ---------------------------------------------------------------------------
__global__ __launch_bounds__(256) void k_embed(
    const float* __restrict__ X, const float* __restrict__ S,
    const float* __restrict__ W1x, const float* __restrict__ W1s,
    float* __restrict__ Xe, float* __restrict__ Se) {
    __shared__ float w1x[256], w1s[256];
    int t = threadIdx.x;
    w1x[t] = W1x[t];
    w1s[t] = W1s[t];
    __syncthreads();
    int gid = blockIdx.x * 256 + t;              // 0 .. 4*4096-1
    int n = gid & (NN - 1);
    int b = gid >> 12;
    const float* Xb = X + (size_t)b * LL * NN + n;
    const float* Sb = S + (size_t)b * LL * NN + n;
    float sv[LL], xv[LL];
#pragma unroll
    for (int l = 0; l < LL; ++l) { sv[l] = Sb[l * NN]; xv[l] = Xb[l * NN]; }
    float* Xeb = Xe + (size_t)b * LL * NN + n;
    float* Seb = Se + (size_t)b * LL * NN + n;
#pragma unroll
    for (int m = 0; m < LL; ++m) {
        float ax = 0.f, as = 0.f;
#pragma unroll
        for (int l = 0; l < LL; ++l) {
            ax += sv[l] * w1x[l * 16 + m];
            as += xv[l] * w1s[l * 16 + m];
        }
        Xeb[m * NN] = ax;
        Seb[m * NN] = as;
    }
}

// ---------------------------------------------------------------------------
// Kernel 2: per (branch,b,n-tile of 16): row max / reciprocal row sum(exp)
// of G = E^T E. One wave per n-tile; G tiles via fp32 WMMA; online softmax;
// butterfly reduction within each 16-lane half. stats[2n]=max, stats[2n+1]=1/sum.
// ---------------------------------------------------------------------------
__global__ __launch_bounds__(256) void k_softmax_stats(
    const float* __restrict__ Exe, const float* __restrict__ Ese,
    float* __restrict__ statsX, float* __restrict__ statsS) {
    int warp = threadIdx.x >> 5, lane = threadIdx.x & 31;
    int wid = blockIdx.x * 8 + warp;             // 0 .. 2047
    int branch = wid >> 10;
    int b = (wid >> 8) & 3;
    int n0 = (wid & 255) * 16;
    const float* E = (branch ? Ese : Exe) + (size_t)b * LL * NN;
    float* stats = (branch ? statsS : statsX) + (size_t)b * NN * 2;
    int h = lane >> 4, col = lane & 15;

    // A: rows = n (lane&15), k = l ; chunk c covers l = 4c..4c+3
    v2f a[4];
#pragma unroll
    for (int c = 0; c < 4; ++c)
#pragma unroll
        for (int v = 0; v < 2; ++v)
            a[c][v] = E[(4 * c + 2 * h + v) * NN + n0 + col];

    float rm[8], rs[8];
#pragma unroll
    for (int v = 0; v < 8; ++v) { rm[v] = -__builtin_inff(); rs[v] = 0.f; }

    for (int m0 = 0; m0 < NN; m0 += 16) {
        v2f bt[4];
#pragma unroll
        for (int c = 0; c < 4; ++c)
#pragma unroll
            for (int v = 0; v < 2; ++v)
                bt[c][v] = E[(4 * c + 2 * h + v) * NN + m0 + col];
        v8f g = {};
#pragma unroll
        for (int c = 0; c < 4; ++c) g = wmma4(a[c], bt[c], g);
#pragma unroll
        for (int v = 0; v < 8; ++v) {
            float gv = g[v];
            float nm = fmaxf(rm[v], gv);
            rs[v] = rs[v] * __expf(rm[v] - nm) + __expf(gv - nm);
            rm[v] = nm;
        }
    }
    // reduce across the 16 lanes sharing each row (stay inside the half)
#pragma unroll
    for (int off = 1; off < 16; off <<= 1) {
#pragma unroll
        for (int v = 0; v < 8; ++v) {
            float om = __shfl_xor(rm[v], off, 32);
            float os = __shfl_xor(rs[v], off, 32);
            float nm = fmaxf(rm[v], om);
            rs[v] = rs[v] * __expf(rm[v] - nm) + os * __expf(om - nm);
            rm[v] = nm;
        }
    }
    if (col == 0) {
#pragma unroll
        for (int v = 0; v < 8; ++v) {
            int n = n0 + v + 8 * h;
            v2f st;
            st[0] = rm[v];
            st[1] = 1.0f / rs[v];
            *(v2f*)&stats[2 * n] = st;
        }
    }
}

// ---------------------------------------------------------------------------
// Kernel 3: Ea[l, m-tile] = sum_n E[l,n] * (exp(G[n,m]-max[n]) / d[n]).
// One wave per m-tile; recompute G tiles, exponentiate+scale, relayout P
// through a column-major padded LDS slice (b64 both ways), second WMMA
// accumulates Ea. Epilogue re-packs Ea into per-t planes (b,t,l,K) so the
// W2-mixing kernel gets contiguous b64 A-operand loads.
// ---------------------------------------------------------------------------
__global__ __launch_bounds__(256) void k_attend(
    const float* __restrict__ Exe, const float* __restrict__ Ese,
    const float* __restrict__ statsX, const float* __restrict__ statsS,
    float* __restrict__ EaPX, float* __restrict__ EaPS) {
    __shared__ float pt[8][16 * 20];             // [wave][col*20 + n], col-major P
    int warp = threadIdx.x >> 5, lane = threadIdx.x & 31;
    int wid = blockIdx.x * 8 + warp;             // 0 .. 2047
    int branch = wid >> 10;
    int b = (wid >> 8) & 3;
    int m0 = (wid & 255) * 16;
    const float* E = (branch ? Ese : Exe) + (size_t)b * LL * NN;
    const float* stats = (branch ? statsS : statsX) + (size_t)b * NN * 2;
    float* EaP = (branch ? EaPS : EaPX);
    int h = lane >> 4, col = lane & 15;

    // fixed B of the Gram tile: k = l, cols = m-tile
    v2f bf[4];
#pragma unroll
    for (int c = 0; c < 4; ++c)
#pragma unroll
        for (int v = 0; v < 2; ++v)
            bf[c][v] = E[(4 * c + 2 * h + v) * NN + m0 + col];

    float* myp = &pt[warp][col * 20];
    v8f acc = {};
    for (int n0 = 0; n0 < NN; n0 += 16) {
        v2f ag[4];
#pragma unroll
        for (int c = 0; c < 4; ++c)
#pragma unroll
            for (int v = 0; v < 2; ++v)
                ag[c][v] = E[(4 * c + 2 * h + v) * NN + n0 + col];
        v8f g = {};
#pragma unroll
        for (int c = 0; c < 4; ++c) g = wmma4(ag[c], bf[c], g);
        // P' = exp(G - rowmax) * (1/rowsum); D-layout -> LDS (column-major)
#pragma unroll
        for (int w2 = 0; w2 < 4; ++w2) {
            v2f p;
#pragma unroll
            for (int u = 0; u < 2; ++u) {
                int v = 2 * w2 + u;
                v2f st = *(const v2f*)&stats[2 * (n0 + v + 8 * h)];
                p[u] = __expf(g[v] - st[0]) * st[1];
            }
            *(v2f*)&myp[8 * h + 2 * w2] = p;     // rows n = 8h+2w2+{0,1}
        }
        v2f a2[4], b2[4];
#pragma unroll
        for (int c = 0; c < 4; ++c) {
            a2[c] = *(const v2f*)&E[col * NN + n0 + 4 * c + 2 * h]; // row l = col
            b2[c] = *(const v2f*)&myp[4 * c + 2 * h];
        }
#pragma unroll
        for (int c = 0; c < 4; ++c) acc = wmma4(a2[c], b2[c], acc);
    }
    // store packed: EaP[((b*2+t)*16 + l)*KDIM + k], with m = 2k+t
    int m = m0 + col;
    int t = m & 1, k = m >> 1;
#pragma unroll
    for (int v = 0; v < 8; ++v) {
        int l = v + 8 * h;
        EaP[(((size_t)b * 2 + t) * LL + l) * KDIM + k] = acc[v];
    }
}

// ---------------------------------------------------------------------------
// Kernel 4: M[b,l,j,t] = sum_k EaP[b,t,l,k]*W2[k,j]; fuse gating, concat,
// and positional encoding (S branch). One wave per (branch,b,t,j-tile).
// ---------------------------------------------------------------------------
__global__ __launch_bounds__(256) void k_mix(
    const float* __restrict__ EaPX, const float* __restrict__ EaPS,
    const float* __restrict__ W2x, const float* __restrict__ W2s,
    const float* __restrict__ X, const float* __restrict__ S,
    float* __restrict__ Xhat, float* __restrict__ Shat) {
    int warp = threadIdx.x >> 5, lane = threadIdx.x & 31;
    int wid = blockIdx.x * 8 + warp;             // 0 .. 2047
    int branch = wid >> 10;
    int rem = wid & 1023;
    int b = rem >> 8;
    int t = (rem >> 7) & 1;
    int j0 = (rem & 127) * 16;
    const float* EaPt = (branch ? EaPS : EaPX) + ((size_t)b * 2 + t) * LL * KDIM;
    const float* W2 = branch ? W2s : W2x;
    const float* In = (branch ? S : X) + (size_t)b * LL * NN;
    float* Hat = branch ? Shat : Xhat;
    int h = lane >> 4, col = lane & 15;

    v8f acc = {};
    for (int k0 = 0; k0 < KDIM; k0 += 16) {
        v2f a[4], bt[4];
#pragma unroll
        for (int c = 0; c < 4; ++c) {
            a[c] = *(const v2f*)&EaPt[col * KDIM + k0 + 4 * c + 2 * h]; // row l = col
#pragma unroll
            for (int v = 0; v < 2; ++v)
                bt[c][v] = W2[(k0 + 4 * c + 2 * h + v) * KDIM + j0 + col];
        }
#pragma unroll
        for (int c = 0; c < 4; ++c) acc = wmma4(a[c], bt[c], acc);
    }
    int j = j0 + col;
    float pe = (t == 0) ? __sinf((float)j) : __cosf((float)j);
#pragma unroll
    for (int v = 0; v < 8; ++v) {
        int l = v + 8 * h;
        float iv = In[l * NN + 2 * j + t];
        float mprod = acc[v] * iv;
        size_t o1 = ((size_t)(b * 32 + l) * KDIM + j) * 2 + t;
        size_t o2 = ((size_t)(b * 32 + l + 16) * KDIM + j) * 2 + t;
        if (branch) { Hat[o1] = mprod + pe; Hat[o2] = iv + pe; }
        else        { Hat[o1] = mprod;      Hat[o2] = iv; }
    }
}

// ---------------------------------------------------------------------------
// Kernel 5: direct 3x3 SAME conv on (4,32,2048,2), both branches -> d_out.
// ---------------------------------------------------------------------------
__global__ __launch_bounds__(256) void k_conv(
    const float* __restrict__ Xhat, const float* __restrict__ Shat,
    const float* __restrict__ Wx, const float* __restrict__ Bx,
    const float* __restrict__ Ws, const float* __restrict__ Bs,
    float* __restrict__ out) {
    int i = blockIdx.x * 256 + threadIdx.x;      // (b,64,2048,2) flat
    int t = i & 1;
    int k = (i >> 1) & 2047;
    int c = (i >> 12) & 63;
    int b = i >> 18;
    int branch = c >> 5;
    int oc = c & 31;
    const float* In = branch ? Shat : Xhat;
    const float* W = branch ? Ws : Wx;
    float acc = branch ? Bs[oc] : Bx[oc];
    const float* Inb = In + (size_t)b * 32 * KDIM * 2;
    const float* Woc = W + oc * 32 * 9;
    for (int ic = 0; ic < 32; ++ic) {
        const float* Ic = Inb + (size_t)ic * KDIM * 2;
        const float* Wic = Woc + ic * 9;
#pragma unroll
        for (int kh = 0; kh < 3; ++kh) {
            int hh = k + kh - 1;
            if ((unsigned)hh < (unsigned)KDIM) {
#pragma unroll
                for (int kw = 0; kw < 3; ++kw) {
                    int w = t + kw - 1;
                    if ((unsigned)w < 2u)
                        acc += Ic[hh * 2 + w] * Wic[kh * 3 + kw];
                }
            }
        }
    }
    out[i] = acc;
}

// ---------------------------------------------------------------------------
extern "C" void kernel_launch(void* const* d_in, const int* in_sizes, int n_in,
                              void* d_out, int out_size, void* d_ws, size_t ws_size,
                              hipStream_t stream) {
    const float* X   = (const float*)d_in[0];
    const float* S   = (const float*)d_in[1];
    const float* W1x = (const float*)d_in[2];
    const float* W1s = (const float*)d_in[3];
    const float* W2x = (const float*)d_in[4];
    const float* W2s = (const float*)d_in[5];
    const float* Cwx = (const float*)d_in[6];
    const float* Cbx = (const float*)d_in[7];
    const float* Cws = (const float*)d_in[8];
    const float* Cbs = (const float*)d_in[9];
    float* out = (float*)d_out;

    float* w = (float*)d_ws;
    float* Xe     = w;                 // 4*16*4096
    float* Se     = Xe + 262144;
    float* EaPX   = Se + 262144;       // packed (b,t,l,K): 4*2*16*2048
    float* EaPS   = EaPX + 262144;
    float* statsX = EaPS + 262144;     // (b, n, {max, 1/sum}): 4*4096*2
    float* statsS = statsX + 32768;
    float* Xhat   = statsS + 32768;    // 4*32*4096
    float* Shat   = Xhat + 524288;

    k_embed<<<64, 256, 0, stream>>>(X, S, W1x, W1s, Xe, Se);
    k_softmax_stats<<<256, 256, 0, stream>>>(Xe, Se, statsX, statsS);
    k_attend<<<256, 256, 0, stream>>>(Xe, Se, statsX, statsS, EaPX, EaPS);
    k_mix<<<256, 256, 0, stream>>>(EaPX, EaPS, W2x, W2s, X, S, Xhat, Shat);
    k_conv<<<4096, 256, 0, stream>>>(Xhat, Shat, Cwx, Cbx, Cws, Cbs, out);
}